// FeatureGrid_73031623901832
// MI455X (gfx1250) — compile-verified
//
#include <hip/hip_runtime.h>
#include <math.h>

typedef float v2f __attribute__((ext_vector_type(2)));
typedef float v8f __attribute__((ext_vector_type(8)));

#define CC 64
#define HH 128
#define WWID 128
#define QTOT 8192
#define TWO_PI 6.28318530717958647692f

__device__ __forceinline__ int iclamp(int v, int lo, int hi) {
    return v < lo ? lo : (v > hi ? hi : v);
}

// ---------------- prep: fold layernorm gains/biases into Wkv / Wq ----------------
// ws layout (floats):
//   [0      .. 8191 ] W1  = diag(ln_gc) @ Wkv        (64 x 128)
//   [8192   .. 8319 ] s1[n]   = sum_c W1[c][n]
//   [8320   .. 8447 ] bvec[n] = sum_c ln_bc[c]*Wkv[c][n]
//   [8448   .. 12543] W2  = diag(ln_gx) @ Wq         (64 x 64)
//   [12544  .. 12607] s2[n]   = sum_c W2[c][n]
//   [12608  .. 12671] b2[n]   = sum_c ln_bx[c]*Wq[c][n]
__global__ __launch_bounds__(256) void fg_prep(
    const float* __restrict__ ln_gx, const float* __restrict__ ln_bx,
    const float* __restrict__ ln_gc, const float* __restrict__ ln_bc,
    const float* __restrict__ Wq,    const float* __restrict__ Wkv,
    float* __restrict__ ws)
{
    int t = threadIdx.x;
    if (t < 128) {
        int n = t;
        float s = 0.f, bv = 0.f;
        #pragma unroll 4
        for (int c = 0; c < CC; ++c) {
            float wv = Wkv[c * 128 + n];
            float w1 = ln_gc[c] * wv;
            ws[c * 128 + n] = w1;
            s  += w1;
            bv += ln_bc[c] * wv;
        }
        ws[8192 + n] = s;
        ws[8320 + n] = bv;
    } else if (t < 192) {
        int n = t - 128;
        float s = 0.f, b2 = 0.f;
        #pragma unroll 4
        for (int c = 0; c < CC; ++c) {
            float wv = Wq[c * 64 + n];
            float w2 = ln_gx[c] * wv;
            ws[8448 + c * 64 + n] = w2;
            s  += w2;
            b2 += ln_bx[c] * wv;
        }
        ws[12544 + n] = s;
        ws[12608 + n] = b2;
    }
}

// ---------------- main: one wave32 per 4 queries ----------------
__global__ __launch_bounds__(256) void fg_main(
    const float* __restrict__ image,  const float* __restrict__ lat,
    const float* __restrict__ coords, const float* __restrict__ Wo,
    const float* __restrict__ bo,     const float* __restrict__ ws,
    float* __restrict__ out)
{
    __shared__ float W1s[CC * 128];     // 32 KB: folded Wkv, B-matrix source
    __shared__ float wscr[8 * 160];     // per-wave scratch: qf[64] | out[64] | murs[32]

    const float* W1g = ws;
    const float* s1g = ws + 8192;
    const float* bvg = ws + 8320;
    const float* W2g = ws + 8448;
    const float* s2g = ws + 12544;
    const float* b2g = ws + 12608;

    const int tid = threadIdx.x;
    for (int i = tid; i < CC * 128; i += 256) W1s[i] = W1g[i];
    __syncthreads();

    const int l  = tid & 31;        // lane
    const int w  = tid >> 5;        // wave in block
    const int n  = l & 15;          // column-within-tile / neighbor id
    const int hi = l >> 4;          // half-wave select

    float* qfbuf  = &wscr[w * 160]; // 64 raw query features
    float* outbuf = qfbuf + 64;     // 64 attention outputs
    float* murs   = outbuf + 64;    // 16 x (mu, rs) for neighbors

    // query-invariant per-lane constants
    float s1p[8], bvp[8];
    #pragma unroll
    for (int t = 0; t < 8; ++t) { s1p[t] = s1g[16 * t + n]; bvp[t] = bvg[16 * t + n]; }
    float s2p[4], b2p[4];
    #pragma unroll
    for (int t = 0; t < 4; ++t) { s2p[t] = s2g[16 * t + n]; b2p[t] = b2g[16 * t + n]; }
    const float bo1 = bo[l], bo2 = bo[l + 32];

    const int qbase = (blockIdx.x * 8 + w) * 4;
    __builtin_prefetch(coords + 2 * qbase, 0, 1);   // global_prefetch_b8

    for (int qi = 0; qi < 4; ++qi) {
        const int q = qbase + qi;
        __builtin_amdgcn_wave_barrier();   // protect wave-scratch reuse across queries

        const float qy = coords[2 * q + 0];
        const float qx = coords[2 * q + 1];
        // grid-index space: grid point i sits exactly at index i
        const float fy = (qy + 1.f) * 64.f - 0.5f;
        const float fx = (qx + 1.f) * 64.f - 0.5f;
        const float y0f = floorf(fy), x0f = floorf(fx);
        const float wy = fy - y0f,   wx = fx - x0f;
        const int y0 = (int)y0f, x0 = (int)x0f;

        const float w00 = (1.f - wy) * (1.f - wx), w01 = (1.f - wy) * wx;
        const float w10 = wy * (1.f - wx),         w11 = wy * wx;
        const float my0 = (y0 >= 0 && y0 < HH)         ? 1.f : 0.f;
        const float my1 = (y0 + 1 >= 0 && y0 + 1 < HH) ? 1.f : 0.f;
        const float mx0 = (x0 >= 0 && x0 < WWID)         ? 1.f : 0.f;
        const float mx1 = (x0 + 1 >= 0 && x0 + 1 < WWID) ? 1.f : 0.f;
        const int yc0 = iclamp(y0, 0, HH - 1),   yc1 = iclamp(y0 + 1, 0, HH - 1);
        const int xc0 = iclamp(x0, 0, WWID - 1), xc1 = iclamp(x0 + 1, 0, WWID - 1);
        const float c00 = w00 * my0 * mx0, c01 = w01 * my0 * mx1;
        const float c10 = w10 * my1 * mx0, c11 = w11 * my1 * mx1;

        // ---- bilinear sample of latents: 2 channels per lane ----
        float qf0, qf1;
        {
            const float* p = lat + (2 * l) * (HH * WWID);
            qf0 = c00 * p[yc0 * WWID + xc0] + c01 * p[yc0 * WWID + xc1]
                + c10 * p[yc1 * WWID + xc0] + c11 * p[yc1 * WWID + xc1];
            p += HH * WWID;
            qf1 = c00 * p[yc0 * WWID + xc0] + c01 * p[yc0 * WWID + xc1]
                + c10 * p[yc1 * WWID + xc0] + c11 * p[yc1 * WWID + xc1];
        }
        qfbuf[2 * l]     = qf0;
        qfbuf[2 * l + 1] = qf1;

        // ---- bilinear sample of input image (1 channel, uniform) ----
        const float qinp = c00 * image[yc0 * WWID + xc0] + c01 * image[yc0 * WWID + xc1]
                         + c10 * image[yc1 * WWID + xc0] + c11 * image[yc1 * WWID + xc1];

        // ---- bilinear sample of analytic coordinate grid (masked, like reference) ----
        const float ysv0 = (2.f * yc0 + 1.f) * (1.f / HH) - 1.f;
        const float ysv1 = (2.f * yc1 + 1.f) * (1.f / HH) - 1.f;
        const float xsv0 = (2.f * xc0 + 1.f) * (1.f / WWID) - 1.f;
        const float xsv1 = (2.f * xc1 + 1.f) * (1.f / WWID) - 1.f;
        const float qcy = c00 * ysv0 + c01 * ysv0 + c10 * ysv1 + c11 * ysv1;
        const float qcx = c00 * xsv0 + c01 * xsv1 + c10 * xsv0 + c11 * xsv1;

        // ---- 16 nearest lattice points = 4x4 window shifted in-bounds ----
        const int by = iclamp((int)y0f - 1, 0, HH - 4);
        const int bx = iclamp((int)x0f - 1, 0, WWID - 4);
        const int ym = by + (n >> 2);
        const int xm = bx + (n & 3);

        // gather the 32 raw channel values this lane's A-fragments need:
        // A[m][c] with c = 4*kk + 2*hi + r  (WMMA f32 16x16x4 A layout)
        float araw[32];
        const float* pg = lat + ym * WWID + xm;
        #pragma unroll
        for (int kk = 0; kk < 16; ++kk) {
            const int c0 = 4 * kk + 2 * hi;
            araw[2 * kk]     = pg[c0 * (HH * WWID)];
            araw[2 * kk + 1] = pg[(c0 + 1) * (HH * WWID)];
        }

        // ---- neighbor LN stats (lane pair covers all 64 channels) ----
        float s = 0.f, sq = 0.f;
        #pragma unroll
        for (int i = 0; i < 32; ++i) { s += araw[i]; sq += araw[i] * araw[i]; }
        s  += __shfl_xor(s, 16, 32);
        sq += __shfl_xor(sq, 16, 32);
        const float mu_m = s * (1.f / 64.f);
        const float var_m = sq * (1.f / 64.f) - mu_m * mu_m;
        const float rs_m = rsqrtf(var_m + 1e-5f);
        if (hi == 0) { murs[2 * n] = mu_m; murs[2 * n + 1] = rs_m; }
        __builtin_amdgcn_wave_barrier();

        // ---- query-feature LN stats (full-wave reduction) ----
        float sx = qf0 + qf1, sqx = qf0 * qf0 + qf1 * qf1;
        #pragma unroll
        for (int m = 1; m < 32; m <<= 1) {
            sx  += __shfl_xor(sx,  m, 32);
            sqx += __shfl_xor(sqx, m, 32);
        }
        const float mux = sx * (1.f / 64.f);
        const float rsx = rsqrtf(sqx * (1.f / 64.f) - mux * mux + 1e-5f);

        // ---- WMMA: raw(16x64) @ W1(64x128), fp32, 8 N-tiles ----
        v8f acc[8];
        const v8f vz = {0.f, 0.f, 0.f, 0.f, 0.f, 0.f, 0.f, 0.f};
        #pragma unroll
        for (int t = 0; t < 8; ++t) acc[t] = vz;

        #pragma unroll
        for (int kk = 0; kk < 16; ++kk) {
            v2f a;
            a.x = araw[2 * kk];
            a.y = araw[2 * kk + 1];
            const int row = 4 * kk + 2 * hi;
            const float* b0 = &W1s[row * 128 + n];
            #pragma unroll
            for (int t = 0; t < 8; ++t) {
                v2f b;
                b.x = b0[16 * t];
                b.y = b0[16 * t + 128];
                acc[t] = __builtin_amdgcn_wmma_f32_16x16x4_f32(
                    false, a, false, b, (short)0, acc[t], false, false);
            }
        }

        // per-row LN fixup values at C/D layout (rows r + 8*hi)
        float muv[8], rsv[8];
        #pragma unroll
        for (int r = 0; r < 8; ++r) {
            muv[r] = murs[2 * (r + 8 * hi)];
            rsv[r] = murs[2 * (r + 8 * hi) + 1];
        }

        // ---- q_vec[d] for d = 16t + n  (folded-LN Wq) ----
        float qv[4] = {0.f, 0.f, 0.f, 0.f};
        for (int c = 0; c < CC; ++c) {
            const float qc = qfbuf[c];
            #pragma unroll
            for (int t = 0; t < 4; ++t) qv[t] += qc * W2g[c * 64 + 16 * t + n];
        }
        #pragma unroll
        for (int t = 0; t < 4; ++t) qv[t] = rsx * (qv[t] - mux * s2p[t]) + b2p[t];

        // ---- sim partials, reduce across the 16 n-lanes ----
        float p[8];
        #pragma unroll
        for (int r = 0; r < 8; ++r) {
            float a = 0.f;
            #pragma unroll
            for (int t = 0; t < 4; ++t) {
                const float kval = rsv[r] * (acc[t][r] - muv[r] * s1p[t]) + bvp[t];
                a += qv[t] * kval;
            }
            p[r] = a;
        }
        #pragma unroll
        for (int m = 1; m < 16; m <<= 1) {
            #pragma unroll
            for (int r = 0; r < 8; ++r) p[r] += __shfl_xor(p[r], m, 32);
        }
        float po[8];
        #pragma unroll
        for (int r = 0; r < 8; ++r) po[r] = __shfl_xor(p[r], 16, 32);

        // ---- softmax over the 16 neighbors ----
        const float scale = 0.125f;   // C^-0.5
        float mx = -1e30f;
        #pragma unroll
        for (int r = 0; r < 8; ++r) {
            p[r]  *= scale;
            po[r] *= scale;
            mx = fmaxf(mx, fmaxf(p[r], po[r]));
        }
        float eA[8], Z = 0.f;
        #pragma unroll
        for (int r = 0; r < 8; ++r) {
            eA[r] = __expf(p[r] - mx);
            Z += eA[r] + __expf(po[r] - mx);
        }
        const float invZ = 1.f / Z;

        // ---- out[d] = attn @ v, d = 16t + n ----
        float oo[4] = {0.f, 0.f, 0.f, 0.f};
        #pragma unroll
        for (int r = 0; r < 8; ++r) {
            #pragma unroll
            for (int t = 0; t < 4; ++t) {
                const float vval = rsv[r] * (acc[4 + t][r] - muv[r] * s1p[4 + t]) + bvp[4 + t];
                oo[t] += eA[r] * vval;
            }
        }
        #pragma unroll
        for (int t = 0; t < 4; ++t) {
            oo[t] += __shfl_xor(oo[t], 16, 32);
            oo[t] *= invZ;
        }
        if (hi == 0) {
            #pragma unroll
            for (int t = 0; t < 4; ++t) outbuf[16 * t + n] = oo[t];
        }
        __builtin_amdgcn_wave_barrier();

        // ---- res = out @ Wo + bo : lane computes columns l and l+32 ----
        float r1 = bo1, r2 = bo2;
        for (int d = 0; d < CC; ++d) {
            const float ov = outbuf[d];
            r1 += ov * Wo[d * 64 + l];
            r2 += ov * Wo[d * 64 + l + 32];
        }

        // ---- write 97-float output row ----
        float* orow = out + (long)q * 97;
        orow[l]      = r1;
        orow[l + 32] = r2;
        if (l == 0) orow[64] = qinp;
        {
            // pe layout: [y: sin0..7,cos0..7, x: sin0..7,cos0..7]; lane l -> element l
            const float x01 = ((hi ? qcx : qcy) + 1.f) * 0.5f;
            const int oct = n & 7;
            const float ang = x01 * (float)(1 << oct) * TWO_PI;
            orow[65 + l] = (n < 8) ? __sinf(ang) : __cosf(ang);
        }
    }
}

extern "C" void kernel_launch(void* const* d_in, const int* in_sizes, int n_in,
                              void* d_out, int out_size, void* d_ws, size_t ws_size,
                              hipStream_t stream) {
    const float* image  = (const float*)d_in[0];
    const float* lat    = (const float*)d_in[1];
    const float* coords = (const float*)d_in[2];
    const float* ln_gx  = (const float*)d_in[3];
    const float* ln_bx  = (const float*)d_in[4];
    const float* ln_gc  = (const float*)d_in[5];
    const float* ln_bc  = (const float*)d_in[6];
    const float* Wq     = (const float*)d_in[7];
    const float* Wkv    = (const float*)d_in[8];
    const float* Wo     = (const float*)d_in[9];
    const float* bo     = (const float*)d_in[10];
    float* ws  = (float*)d_ws;
    float* out = (float*)d_out;

    fg_prep<<<1, 256, 0, stream>>>(ln_gx, ln_bx, ln_gc, ln_bc, Wq, Wkv, ws);
    fg_main<<<256, 256, 0, stream>>>(image, lat, coords, Wo, bo, ws, out);
}